// JasThreeBody_45749991637048
// MI455X (gfx1250) — compile-verified
//
#include <hip/hip_runtime.h>

typedef float v2f __attribute__((ext_vector_type(2)));
typedef float v8f __attribute__((ext_vector_type(8)));

#define NSAMP      512
#define NFEAT      128
#define LDS_STRIDE 132          // pad 128 -> 132 floats: lane rows hit distinct LDS banks
#define NTRI       341376      // 128*127*126/6, exactly divisible by 8
#define TRI_Y      16           // triple-dimension split (grid.y)
#define WAVES_PB   8            // 256 threads = 8 wave32

__global__ void JasThreeBody_zero_kernel(float* __restrict__ out) {
    int t = blockIdx.x * blockDim.x + threadIdx.x;
    if (t < NSAMP) out[t] = 0.0f;
}

__global__ __launch_bounds__(256)
void JasThreeBody_45749991637048_kernel(const float* __restrict__ x,
                                        const float* __restrict__ kern,
                                        const int*   __restrict__ tri,
                                        float*       __restrict__ out) {
    __shared__ float xs[16 * LDS_STRIDE];   // 16 sample rows, padded (8448 B)

    const int tid = threadIdx.x;
    const int sb  = blockIdx.x;             // sample block: samples [sb*16, sb*16+16)

    // Stage 16 contiguous x rows (2048 f32) into padded LDS. Coalesced global reads.
    const float* xsrc = x + sb * (16 * NFEAT);
    for (int idx = tid; idx < 16 * NFEAT; idx += 256) {
        int mm = idx >> 7;             // sample row 0..15
        int ff = idx & (NFEAT - 1);    // feature 0..127
        xs[mm * LDS_STRIDE + ff] = xsrc[idx];
    }
    __syncthreads();

    const int wave = tid >> 5;
    const int lane = tid & 31;
    const int m    = lane & 15;        // sample row (A-matrix M for this lane)
    const int half = lane >> 4;        // 0 -> K={0,1} in VGPR0/1 ; 1 -> K={2,3} (ISA 16x4 f32 A layout)
    const float* __restrict__ xm = xs + m * LDS_STRIDE;

    // Worker id is identical across all 32 lanes of a wave; readfirstlane makes it an
    // SGPR so the triple loop becomes a scalar (s_cbranch) loop: no exec masking, and
    // EXEC is structurally all-ones at every WMMA as the ISA requires.
    const int tw       = __builtin_amdgcn_readfirstlane(blockIdx.y * WAVES_PB + wave);
    const int nworkers = TRI_Y * WAVES_PB;                 // 128 wave-workers over triples

    // Two independent f32 accumulator tiles -> back-to-back WMMAs don't serialize on C.
    v8f acc0 = {0.f, 0.f, 0.f, 0.f, 0.f, 0.f, 0.f, 0.f};
    v8f acc1 = acc0;
    const v2f bones = {1.0f, 1.0f};    // B = all-ones: D[:,col] identical for all col,
                                       // independent of the B VGPR layout.

    // Each wave iteration consumes 8 triples: WMMA q handles c0+4q .. c0+4q+3;
    // lane half h supplies K = {2h, 2h+1} of that tile.
    for (int c0 = tw * 8; c0 < NTRI; c0 += nworkers * 8) {
        const int  cా = c0 + 2 * half;           // base for this lane's pair (q = 0)
        const int  ccA = c0 + 2 * half;
        const int  ccB = ccA + 4;
        const int* tpA = tri + ccA * 3;          // 6 consecutive ints (two triples)
        const int* tpB = tri + ccB * 3;
        const v2f  wA  = *(const v2f*)(kern + ccA);   // cc even -> 8B aligned b64 load
        const v2f  wB  = *(const v2f*)(kern + ccB);

        const int iA0 = tpA[0], jA0 = tpA[1], kA0 = tpA[2];
        const int iA1 = tpA[3], jA1 = tpA[4], kA1 = tpA[5];
        const int iB0 = tpB[0], jB0 = tpB[1], kB0 = tpB[2];
        const int iB1 = tpB[3], jB1 = tpB[4], kB1 = tpB[5];

        // A[m, K] = kernel-weighted triple product (f32, matches reference precision)
        v2f aA, aB;
        aA.x = xm[iA0] * xm[jA0] * xm[kA0] * wA.x;
        aA.y = xm[iA1] * xm[jA1] * xm[kA1] * wA.y;
        aB.x = xm[iB0] * xm[jB0] * xm[kB0] * wB.x;
        aB.y = xm[iB1] * xm[jB1] * xm[kB1] * wB.y;

        acc0 = __builtin_amdgcn_wmma_f32_16x16x4_f32(
            false, aA, false, bones, (short)0, acc0, false, false);
        acc1 = __builtin_amdgcn_wmma_f32_16x16x4_f32(
            false, aB, false, bones, (short)0, acc1, false, false);
        (void)cా;
    }

    const v8f acc = acc0 + acc1;

    // C/D layout: VGPR r, lanes 0-15 -> M=r ; lanes 16-31 -> M=8+r. Lane 0 holds
    // M=0..7, lane 16 holds M=8..15 (every column equal since B = ones).
    if (m == 0) {
        float* o = out + sb * 16 + half * 8;
#pragma unroll
        for (int r = 0; r < 8; ++r)
            atomicAdd(&o[r], acc[r]);
    }
}

extern "C" void kernel_launch(void* const* d_in, const int* in_sizes, int n_in,
                              void* d_out, int out_size, void* d_ws, size_t ws_size,
                              hipStream_t stream) {
    (void)in_sizes; (void)n_in; (void)d_ws; (void)ws_size; (void)out_size;
    const float* x    = (const float*)d_in[0];   // (512, 128) f32
    const float* kern = (const float*)d_in[1];   // (341376,)  f32
    const int*   tri  = (const int*)  d_in[2];   // (341376,3) i32
    float*       out  = (float*)d_out;           // (512,)     f32

    // Zero output (harness poisons it), then accumulate via atomics. Same stream -> ordered.
    JasThreeBody_zero_kernel<<<dim3(2), dim3(256), 0, stream>>>(out);
    JasThreeBody_45749991637048_kernel<<<dim3(NSAMP / 16, TRI_Y), dim3(256), 0, stream>>>(
        x, kern, tri, out);
}